// LearnableMatrixNMS_69741678952499
// MI455X (gfx1250) — compile-verified
//
#include <hip/hip_runtime.h>
#include <math.h>

#ifndef __has_builtin
#define __has_builtin(x) 0
#endif

#if __has_builtin(__builtin_amdgcn_global_load_async_to_lds_b128)
#define HAS_ASYNC_B128 1
#else
#define HAS_ASYNC_B128 0
#endif
#if __has_builtin(__builtin_amdgcn_global_load_async_to_lds_b32)
#define HAS_ASYNC_B32 1
#else
#define HAS_ASYNC_B32 0
#endif

static constexpr int kB = 16;      // batch
static constexpr int kN = 2134;    // boxes
static constexpr int kTILE = 256;  // columns (j) per block
static constexpr int kCHUNK = 512; // rows (i) per block
static constexpr int kNCH = (kN + kCHUNK - 1) / kCHUNK;  // 5

typedef __attribute__((__vector_size__(4 * sizeof(int)))) int v4i;

// ---------- CDNA5 async global->LDS staging (ASYNCcnt path) ----------
__device__ __forceinline__ void async_copy_b128(void* lds_dst, const void* g_src) {
#if HAS_ASYNC_B128
  __builtin_amdgcn_global_load_async_to_lds_b128(
      (__attribute__((address_space(1))) v4i*)(g_src),
      (__attribute__((address_space(3))) v4i*)(lds_dst), /*off=*/0, /*cpol=*/0);
#else
  *reinterpret_cast<float4*>(lds_dst) = *reinterpret_cast<const float4*>(g_src);
#endif
}

__device__ __forceinline__ void async_copy_b32(void* lds_dst, const void* g_src) {
#if HAS_ASYNC_B32
  __builtin_amdgcn_global_load_async_to_lds_b32(
      (__attribute__((address_space(1))) int*)(g_src),
      (__attribute__((address_space(3))) int*)(lds_dst), /*off=*/0, /*cpol=*/0);
#else
  *reinterpret_cast<int*>(lds_dst) = *reinterpret_cast<const int*>(g_src);
#endif
}

__device__ __forceinline__ void async_join_and_barrier() {
#if HAS_ASYNC_B128 || HAS_ASYNC_B32
#if __has_builtin(__builtin_amdgcn_s_wait_asynccnt)
  __builtin_amdgcn_s_wait_asynccnt(0);
#else
  asm volatile("s_wait_asynccnt 0" ::: "memory");
#endif
#endif
  __syncthreads();
}

// IoU with precomputed areas; v_rcp_f32 (TRANS, co-executes) instead of IEEE div
__device__ __forceinline__ float iou_pair(float4 bi, float ai, float4 bj, float aj) {
  float lx = fmaxf(bi.x, bj.x);
  float ly = fmaxf(bi.y, bj.y);
  float rx = fminf(bi.z, bj.z);
  float ry = fminf(bi.w, bj.w);
  float w = fmaxf(rx - lx, 0.0f);
  float h = fmaxf(ry - ly, 0.0f);
  float inter = w * h;
  float uni = (ai + aj) - inter;
  return inter * __builtin_amdgcn_rcpf(uni + 1e-12f);
}

// K0: zero the atomic accumulators (workspace is poisoned with 0xAA).
__global__ void k_zero(float* __restrict__ p, int count) {
  int i = blockIdx.x * blockDim.x + threadIdx.x;
  if (i < count) p[i] = 0.0f;
}

// K1: stable descending rank by score (tie: lower index first) == argsort(-s);
// scatter box + area into sorted order.
__global__ void k_rank_gather(const float* __restrict__ boxes,   // [B,4,N]
                              const float* __restrict__ scores,  // [B,1,N]
                              float4* __restrict__ sboxes,       // [B,N]
                              float* __restrict__ sarea,         // [B,N]
                              int* __restrict__ rank) {          // [B,N]
  const int b = blockIdx.x;
  __shared__ float s[kN];
  const float* sc = scores + b * kN;
  for (int i = threadIdx.x; i < kN; i += blockDim.x) s[i] = sc[i];
  __syncthreads();

  const int n = blockIdx.y * blockDim.x + threadIdx.x;
  if (n >= kN) return;
  const float my = s[n];
  int r = 0;
  for (int i = 0; i < kN; ++i) {
    float v = s[i];
    r += (v > my) || (v == my && i < n);
  }
  rank[b * kN + n] = r;
  const float* bb = boxes + (size_t)b * 4 * kN;
  float x1 = bb[n], y1 = bb[kN + n], x2 = bb[2 * kN + n], y2 = bb[3 * kN + n];
  sboxes[b * kN + r] = make_float4(x1, y1, x2, y2);
  sarea[b * kN + r] = (x2 - x1) * (y2 - y1);
}

// K2: partial column-max of IoU over i-chunk [i0,iend), folded with u32
// atomicMax on float bits (valid: IoU >= 0 -> bit order == float order).
__global__ void k_colmax_part(const float4* __restrict__ sboxes,
                              const float* __restrict__ sarea,
                              unsigned int* __restrict__ ioumax_bits) {
  const int b = blockIdx.x;
  const int j0 = blockIdx.y * kTILE;
  const int i0 = blockIdx.z * kCHUNK;
  const int jmax = min(kN, j0 + kTILE);
  if (i0 >= jmax) return;  // uniform: this chunk is entirely below the triangle
  const int iend = min(jmax, i0 + kCHUNK);
  const int cnt = iend - i0;

  __shared__ float4 lb[kCHUNK];
  __shared__ float la[kCHUNK];
  const float4* srcb = sboxes + (size_t)b * kN + i0;
  const float* srca = sarea + (size_t)b * kN + i0;
  for (int t = threadIdx.x; t < cnt; t += kTILE) {
    async_copy_b128(&lb[t], &srcb[t]);
    async_copy_b32(&la[t], &srca[t]);
  }
  async_join_and_barrier();

  const int j = j0 + (int)threadIdx.x;
  if (j >= kN) return;
  const int myend = min(iend, j);
  if (myend <= i0) return;
  const float4 bj = sboxes[(size_t)b * kN + j];
  const float aj = sarea[(size_t)b * kN + j];
  float m = 0.0f;
  for (int i = i0; i < myend; ++i)
    m = fmaxf(m, iou_pair(lb[i - i0], la[i - i0], bj, aj));
  if (m > 0.0f)
    atomicMax(&ioumax_bits[(size_t)b * kN + j], __float_as_uint(m));
}

// K3: partial dmin[j] = min_{i in chunk, i<j}(ioumax[i]^2 - IoU(i,j)^2),
// clamped to <=0 (true dmin <=0 via row i=0), folded with u32 atomicMax on
// float bits (valid: for x,y<=0, bits(x)>bits(y) <=> x<y; init 0 = identity).
__global__ void k_dmin_part(const float4* __restrict__ sboxes,
                            const float* __restrict__ sarea,
                            const float* __restrict__ ioumax,
                            unsigned int* __restrict__ dmin_bits) {
  const int b = blockIdx.x;
  const int j0 = blockIdx.y * kTILE;
  const int i0 = blockIdx.z * kCHUNK;
  const int jmax = min(kN, j0 + kTILE);
  if (i0 >= jmax) return;
  const int iend = min(jmax, i0 + kCHUNK);
  const int cnt = iend - i0;

  __shared__ float4 lb[kCHUNK];
  __shared__ float la[kCHUNK];
  __shared__ float lm2[kCHUNK];
  const float4* srcb = sboxes + (size_t)b * kN + i0;
  const float* srca = sarea + (size_t)b * kN + i0;
  const float* srcm = ioumax + (size_t)b * kN + i0;
  for (int t = threadIdx.x; t < cnt; t += kTILE) {
    async_copy_b128(&lb[t], &srcb[t]);
    async_copy_b32(&la[t], &srca[t]);
    float mv = srcm[t];
    lm2[t] = mv * mv;
  }
  async_join_and_barrier();

  const int j = j0 + (int)threadIdx.x;
  if (j >= kN) return;
  const int myend = min(iend, j);
  if (myend <= i0) return;
  const float4 bj = sboxes[(size_t)b * kN + j];
  const float aj = sarea[(size_t)b * kN + j];
  float d = 0.0f;  // identity; true dmin <= 0
  for (int i = i0; i < myend; ++i) {
    float io = iou_pair(lb[i - i0], la[i - i0], bj, aj);
    d = fminf(d, lm2[i - i0] - io * io);
  }
  if (d < 0.0f)
    atomicMax(&dmin_bits[(size_t)b * kN + j], __float_as_uint(d));
}

// K4: out[b,n] = sum_m exp(dmin[rank[n]]/(sigma_m+1e-12)) * mw[b,m,n]
__global__ void k_final(const float* __restrict__ mw,    // [B,3,N]
                        const int* __restrict__ rank,    // [B,N]
                        const float* __restrict__ dmin,  // [B,N]
                        float* __restrict__ out) {       // [B,1,N]
  const int idx = blockIdx.x * blockDim.x + threadIdx.x;
  if (idx >= kB * kN) return;
  const int b = idx / kN;
  const int n = idx - b * kN;
  const float d = dmin[b * kN + rank[idx]];
  const float inv0 = 1.0f / (0.1f + 1e-12f);
  const float inv1 = 1.0f / (0.2f + 1e-12f);
  const float inv2 = 1.0f / (0.4f + 1e-12f);
  const float g0 = expf(d * inv0);
  const float g1 = expf(d * inv1);
  const float g2 = expf(d * inv2);
  const float* m = mw + (size_t)b * 3 * kN + n;
  out[idx] = g0 * m[0] + g1 * m[kN] + g2 * m[2 * kN];
}

extern "C" void kernel_launch(void* const* d_in, const int* in_sizes, int n_in,
                              void* d_out, int out_size, void* d_ws, size_t ws_size,
                              hipStream_t stream) {
  (void)in_sizes; (void)n_in; (void)out_size; (void)ws_size;
  const float* mw     = (const float*)d_in[0];  // [B,3,N]
  const float* boxes  = (const float*)d_in[1];  // [B,4,N]
  const float* scores = (const float*)d_in[2];  // [B,1,N]
  float* out = (float*)d_out;                   // [B,1,N] float32

  // workspace layout (ioumax and dmin adjacent -> one zero pass)
  float4* sboxes = (float4*)d_ws;                      // B*N float4
  float*  sarea  = (float*)(sboxes + (size_t)kB * kN); // B*N float
  int*    rank   = (int*)(sarea + (size_t)kB * kN);    // B*N int
  float*  ioumax = (float*)(rank + (size_t)kB * kN);   // B*N float (atomic)
  float*  dmin   = ioumax + (size_t)kB * kN;           // B*N float (atomic)

  k_zero<<<(2 * kB * kN + 255) / 256, 256, 0, stream>>>(ioumax, 2 * kB * kN);

  dim3 g1(kB, (kN + 255) / 256);
  k_rank_gather<<<g1, 256, 0, stream>>>(boxes, scores, sboxes, sarea, rank);

  dim3 g2(kB, (kN + kTILE - 1) / kTILE, kNCH);
  k_colmax_part<<<g2, kTILE, 0, stream>>>(sboxes, sarea, (unsigned int*)ioumax);
  k_dmin_part<<<g2, kTILE, 0, stream>>>(sboxes, sarea, ioumax, (unsigned int*)dmin);

  k_final<<<(kB * kN + 255) / 256, 256, 0, stream>>>(mw, rank, dmin, out);
}